// CRF_77799037600446
// MI455X (gfx1250) — compile-verified
//
#include <hip/hip_runtime.h>
#include <stdint.h>

// ---------------- problem constants ----------------
#define S 7
#define SEQ_LEN (1 << 21)           // 2097152
#define NEG_INF (-10000.0f)

// chunking
#define CHUNK 64                    // steps per chunk
#define NCHUNK (SEQ_LEN / CHUNK)    // 32768
#define CPB 64                      // chunks (== threads) per block in k1/k3/k5
#define NBLK (NCHUNK / CPB)         // 512 blocks
#define STEP_TILE 8                 // steps staged per LDS tile
#define NTILES (CHUNK / STEP_TILE)  // 8

// level-2 scan
#define K2_THREADS 256
#define SUPER (NCHUNK / K2_THREADS) // 128 chunks per k2/k4 thread

// ---------------- workspace layout (bytes) ----------------
#define OFF_CHUNKM 0u                                       // NCHUNK*49 f32
#define OFF_ALPHA  (OFF_CHUNKM + (size_t)NCHUNK * 49 * 4)   // NCHUNK*7 f32
#define OFF_CRUMBS (OFF_ALPHA + (size_t)NCHUNK * 7 * 4)     // SEQ_LEN u32
#define OFF_HMAP   (OFF_CRUMBS + (size_t)SEQ_LEN * 4)       // NCHUNK u32
#define OFF_SEND   (OFF_HMAP + (size_t)NCHUNK * 4)          // NCHUNK u32
#define OFF_MISC   (OFF_SEND + (size_t)NCHUNK * 4)          // best + spare

typedef uint32_t v4u __attribute__((ext_vector_type(4)));
typedef int v8i __attribute__((ext_vector_type(8)));
typedef int v4i __attribute__((ext_vector_type(4)));

// ---------------- CDNA5 async/TDM helpers ----------------
__device__ __forceinline__ uint32_t lds_lo32(const void* p) {
  // generic address of a __shared__ object: low 32 bits == LDS byte offset
  return (uint32_t)(uintptr_t)p;
}
__device__ __forceinline__ void async_store_b128(uint32_t lds_off, void* g) {
  asm volatile("global_store_async_from_lds_b128 %0, %1, off"
               :: "v"((uint64_t)(uintptr_t)g), "v"(lds_off) : "memory");
}
__device__ __forceinline__ void wait_async0() {
  asm volatile("s_wait_asynccnt 0" ::: "memory");
}
__device__ __forceinline__ void wait_ds0() {
  asm volatile("s_wait_dscnt 0" ::: "memory");
}

// Issue one TDM tensor load: 3D tile (X=8 steps contig, Y=64 chunks stride 64,
// Z=7 states stride SEQ_LEN) -> contiguous LDS [state][chunk][step].
__device__ __forceinline__ void tdm_tile_load(const float* em, int cb, int jt,
                                              uint32_t lds_byte_addr) {
  uint64_t ga = (uint64_t)(uintptr_t)(em + (size_t)cb * CHUNK +
                                      (size_t)jt * STEP_TILE);
  // group0: count=1 | lds_addr | global_addr(57b) | type=2 in bits 127:126
  v4u g0 = {1u, lds_byte_addr, (uint32_t)ga,
            (uint32_t)(ga >> 32) | 0x80000000u};
  // group1: data_size=4B(code2); tensor_dim0=SEQ_LEN; tensor_dim1=NCHUNK;
  // tile=(8,64,7); dim0_stride=64; dim1_stride=SEQ_LEN  (element units)
  v8i g1 = {(int)0x20000,
            (int)((SEQ_LEN & 0xFFFFu) << 16),
            (int)((SEQ_LEN >> 16) | ((NCHUNK & 0xFFFFu) << 16)),
            (int)((NCHUNK >> 16) | (8u << 16)),
            (int)(64u | (7u << 16)),
            (int)64u,
            (int)((SEQ_LEN & 0xFFFFu) << 16),
            (int)(SEQ_LEN >> 16)};
  // group2: tensor_dim2=7, higher dims unused
  v4i g2 = {7, 0, 0, 0};
  v4i g3 = {0, 0, 0, 0};
  v8i g4 = {0, 0, 0, 0, 0, 0, 0, 0};  // reserved/extended group: zeros
  __builtin_amdgcn_tensor_load_to_lds(g0, g1, g2, g3, g4, 0);
}

// identity 7->7 map packed 3 bits/state
__device__ __forceinline__ uint32_t idmap() {
  uint32_t g = 0;
#pragma unroll
  for (int s = 0; s < S; ++s) g |= (uint32_t)s << (3 * s);
  return g;
}

// =====================================================================
// k1: per-chunk max-plus matrix P_c (7x7), one thread per chunk.
// Emissions staged by TDM (tensor_load_to_lds), double-buffered.
// =====================================================================
__global__ __launch_bounds__(CPB) void k1_chunk_mats(
    const float* __restrict__ em, const float* __restrict__ trans,
    float* __restrict__ chunkM) {
  __shared__ alignas(16) float ebuf[2][S * CPB * STEP_TILE];
  const int tid = threadIdx.x;
  const int cb = blockIdx.x * CPB;
  const int c = cb + tid;

  float A[S * S];
#pragma unroll
  for (int i = 0; i < S * S; ++i) A[i] = trans[i];

  float P[S * S];
#pragma unroll
  for (int n = 0; n < S; ++n)
#pragma unroll
    for (int p = 0; p < S; ++p) P[n * S + p] = (n == p) ? 0.0f : NEG_INF;

  if (tid < 32) tdm_tile_load(em, cb, 0, lds_lo32(&ebuf[0][0]));

  for (int jt = 0; jt < NTILES; ++jt) {
    const int buf = jt & 1;
    __syncthreads();  // everyone done reading ebuf[buf^1] (prev compute)
    if (tid < 32) {   // wave-uniform: only wave 0 drives the TDM
      if (jt + 1 < NTILES) {
        tdm_tile_load(em, cb, jt + 1, lds_lo32(&ebuf[buf ^ 1][0]));
        __builtin_amdgcn_s_wait_tensorcnt(1);  // in-order: tile jt done
      } else {
        __builtin_amdgcn_s_wait_tensorcnt(0);
      }
    }
    __syncthreads();  // tile jt visible to all waves
#pragma unroll 1
    for (int jj = 0; jj < STEP_TILE; ++jj) {
      float e[S];
#pragma unroll
      for (int s = 0; s < S; ++s)
        e[s] = ebuf[buf][(s * CPB + tid) * STEP_TILE + jj];
      float Q[S * S];
#pragma unroll
      for (int p = 0; p < S; ++p) {
        float col[S];
#pragma unroll
        for (int k = 0; k < S; ++k) col[k] = e[k] + P[k * S + p];
#pragma unroll
        for (int n = 0; n < S; ++n) {
          float m = A[n * S + 0] + col[0];
#pragma unroll
          for (int k = 1; k < S; ++k) m = fmaxf(m, A[n * S + k] + col[k]);
          Q[n * S + p] = m;
        }
      }
#pragma unroll
      for (int i = 0; i < S * S; ++i) P[i] = Q[i];
    }
  }
#pragma unroll
  for (int i = 0; i < S * S; ++i) chunkM[(size_t)c * (S * S) + i] = P[i];
}

// =====================================================================
// k2: hierarchical scan over chunk matrices -> alpha_in per chunk,
// plus final score + argmax state. One block of 256 threads.
// =====================================================================
__global__ __launch_bounds__(K2_THREADS) void k2_scan(
    const float* __restrict__ chunkM, float* __restrict__ alpha_in,
    int* __restrict__ bestp, float* __restrict__ score_out) {
  __shared__ float Qs[K2_THREADS][S * S];
  __shared__ float As[K2_THREADS][S];
  const int tid = threadIdx.x;
  const size_t c0 = (size_t)tid * SUPER;

  float Q[S * S];
#pragma unroll
  for (int n = 0; n < S; ++n)
#pragma unroll
    for (int p = 0; p < S; ++p) Q[n * S + p] = (n == p) ? 0.0f : NEG_INF;
#pragma unroll 1
  for (int cc = 0; cc < SUPER; ++cc) {
    float R[S * S];
#pragma unroll
    for (int i = 0; i < S * S; ++i) R[i] = chunkM[(c0 + cc) * (S * S) + i];
    float N[S * S];
#pragma unroll
    for (int p = 0; p < S; ++p) {
      float col[S];
#pragma unroll
      for (int k = 0; k < S; ++k) col[k] = Q[k * S + p];
#pragma unroll
      for (int n = 0; n < S; ++n) {
        float m = R[n * S + 0] + col[0];
#pragma unroll
        for (int k = 1; k < S; ++k) m = fmaxf(m, R[n * S + k] + col[k]);
        N[n * S + p] = m;
      }
    }
#pragma unroll
    for (int i = 0; i < S * S; ++i) Q[i] = N[i];
  }
#pragma unroll
  for (int i = 0; i < S * S; ++i) Qs[tid][i] = Q[i];
  __syncthreads();

  if (tid == 0) {
    float a[S];
#pragma unroll
    for (int s = 0; s < S; ++s) a[s] = (s == 0) ? 0.0f : NEG_INF;  // START_IX=0
#pragma unroll 1
    for (int i = 0; i < K2_THREADS; ++i) {
#pragma unroll
      for (int s = 0; s < S; ++s) As[i][s] = a[s];
      float na[S];
#pragma unroll
      for (int n = 0; n < S; ++n) {
        float m = Qs[i][n * S + 0] + a[0];
#pragma unroll
        for (int p = 1; p < S; ++p) m = fmaxf(m, Qs[i][n * S + p] + a[p]);
        na[n] = m;
      }
#pragma unroll
      for (int s = 0; s < S; ++s) a[s] = na[s];
    }
    int b = 0;
    float bv = a[0];
#pragma unroll
    for (int s = 1; s < S; ++s)
      if (a[s] > bv) { bv = a[s]; b = s; }
    bestp[0] = b;
    score_out[0] = bv;
  }
  __syncthreads();

  float a[S];
#pragma unroll
  for (int s = 0; s < S; ++s) a[s] = As[tid][s];
#pragma unroll 1
  for (int cc = 0; cc < SUPER; ++cc) {
#pragma unroll
    for (int s = 0; s < S; ++s) alpha_in[(c0 + cc) * S + s] = a[s];
    float R[S * S];
#pragma unroll
    for (int i = 0; i < S * S; ++i) R[i] = chunkM[(c0 + cc) * (S * S) + i];
    float na[S];
#pragma unroll
    for (int n = 0; n < S; ++n) {
      float m = R[n * S + 0] + a[0];
#pragma unroll
      for (int p = 1; p < S; ++p) m = fmaxf(m, R[n * S + p] + a[p]);
      na[n] = m;
    }
#pragma unroll
    for (int s = 0; s < S; ++s) a[s] = na[s];
  }
}

// =====================================================================
// k3: re-scan chunks, emit packed crumbs (7x3 bits/step; staged in LDS,
// written out with async-from-LDS stores) and packed backtrace map H_c.
// Emission tiles come in via TDM.
// =====================================================================
__global__ __launch_bounds__(CPB) void k3_rescan(
    const float* __restrict__ em, const float* __restrict__ trans,
    const float* __restrict__ alpha_in, uint32_t* __restrict__ crumbs,
    uint32_t* __restrict__ hmap) {
  __shared__ alignas(16) float ebuf[2][S * CPB * STEP_TILE];
  __shared__ alignas(16) uint32_t cbuf[2][CPB * STEP_TILE];
  const int tid = threadIdx.x;
  const int cb = blockIdx.x * CPB;
  const int c = cb + tid;

  float A[S * S];
#pragma unroll
  for (int i = 0; i < S * S; ++i) A[i] = trans[i];

  float al[S];
#pragma unroll
  for (int s = 0; s < S; ++s) al[s] = alpha_in[(size_t)c * S + s];
  uint32_t g = idmap();

  if (tid < 32) tdm_tile_load(em, cb, 0, lds_lo32(&ebuf[0][0]));

  for (int jt = 0; jt < NTILES; ++jt) {
    const int buf = jt & 1;
    __syncthreads();  // prev compute done (ebuf[buf^1] reads, cbuf writes)
    if (tid < 32) {
      if (jt + 1 < NTILES) {
        tdm_tile_load(em, cb, jt + 1, lds_lo32(&ebuf[buf ^ 1][0]));
        __builtin_amdgcn_s_wait_tensorcnt(1);
      } else {
        __builtin_amdgcn_s_wait_tensorcnt(0);
      }
    }
    wait_async0();    // crumb stores from earlier tiles done (cbuf reusable)
    __syncthreads();  // tile jt visible
#pragma unroll 1
    for (int jj = 0; jj < STEP_TILE; ++jj) {
      float e[S];
#pragma unroll
      for (int s = 0; s < S; ++s)
        e[s] = ebuf[buf][(s * CPB + tid) * STEP_TILE + jj];
      float na[S];
      uint32_t w = 0;
#pragma unroll
      for (int n = 0; n < S; ++n) {
        float m = al[0] + e[0] + A[n * S + 0];
        int arg = 0;
#pragma unroll
        for (int p = 1; p < S; ++p) {
          float v = al[p] + e[p] + A[n * S + p];
          if (v > m) { m = v; arg = p; }
        }
        na[n] = m;
        w |= (uint32_t)arg << (3 * n);
      }
#pragma unroll
      for (int s = 0; s < S; ++s) al[s] = na[s];
      uint32_t gn = 0;
#pragma unroll
      for (int s = 0; s < S; ++s) {
        uint32_t cr = (w >> (3 * s)) & 7u;
        uint32_t v = (g >> (3 * cr)) & 7u;
        gn |= v << (3 * s);
      }
      g = gn;
      cbuf[buf][tid * STEP_TILE + jj] = w;
    }
    wait_ds0();  // this wave's LDS crumb writes visible to async engine
    {
      uint32_t l0 = lds_lo32(&cbuf[buf][tid * STEP_TILE + 0]);
      uint32_t l1 = lds_lo32(&cbuf[buf][tid * STEP_TILE + 4]);
      uint32_t* gp = crumbs + (size_t)c * CHUNK + jt * STEP_TILE;
      async_store_b128(l0, gp);
      async_store_b128(l1, gp + 4);
    }
  }
  wait_async0();
  hmap[c] = g;
}

// =====================================================================
// k4: hierarchical backward scan over chunk maps -> state at end of
// every chunk. One block of 256 threads.
// =====================================================================
__global__ __launch_bounds__(K2_THREADS) void k4_backscan(
    const uint32_t* __restrict__ hmap, const int* __restrict__ bestp,
    uint32_t* __restrict__ send) {
  __shared__ uint32_t Sm[K2_THREADS];
  __shared__ int Sin[K2_THREADS];
  const int tid = threadIdx.x;
  const size_t c0 = (size_t)tid * SUPER;

  uint32_t M = idmap();
#pragma unroll 1
  for (int cc = SUPER - 1; cc >= 0; --cc) {
    uint32_t h = hmap[c0 + cc];
    uint32_t Mn = 0;
#pragma unroll
    for (int s = 0; s < S; ++s) {
      uint32_t x = (M >> (3 * s)) & 7u;
      uint32_t v = (h >> (3 * x)) & 7u;
      Mn |= v << (3 * s);
    }
    M = Mn;
  }
  Sm[tid] = M;
  __syncthreads();
  if (tid == 0) {
    int s = bestp[0];
#pragma unroll 1
    for (int i = K2_THREADS - 1; i >= 0; --i) {
      Sin[i] = s;
      s = (int)((Sm[i] >> (3 * s)) & 7u);
    }
  }
  __syncthreads();
  int s = Sin[tid];
#pragma unroll 1
  for (int cc = SUPER - 1; cc >= 0; --cc) {
    send[c0 + cc] = (uint32_t)s;
    uint32_t h = hmap[c0 + cc];
    s = (int)((h >> (3 * s)) & 7u);
  }
}

// =====================================================================
// k5: per-chunk backtrace replay -> int32 path.
// =====================================================================
__global__ __launch_bounds__(CPB) void k5_writepath(
    const uint32_t* __restrict__ crumbs, const uint32_t* __restrict__ send,
    int* __restrict__ path) {
  const int c = blockIdx.x * CPB + threadIdx.x;
  int s = (int)send[c];
  const size_t base = (size_t)c * CHUNK;
#pragma unroll 1
  for (int j = CHUNK - 1; j >= 0; --j) {
    path[base + j] = s;
    uint32_t w = crumbs[base + j];
    s = (int)((w >> (3 * s)) & 7u);
  }
}

// =====================================================================
extern "C" void kernel_launch(void* const* d_in, const int* in_sizes, int n_in,
                              void* d_out, int out_size, void* d_ws,
                              size_t ws_size, hipStream_t stream) {
  (void)in_sizes; (void)n_in; (void)ws_size;
  const float* em = (const float*)d_in[0];     // [1,7,SEQ_LEN] f32
  const float* tr = (const float*)d_in[1];     // [7,7] f32
  char* ws = (char*)d_ws;
  float* chunkM = (float*)(ws + OFF_CHUNKM);
  float* alpha_in = (float*)(ws + OFF_ALPHA);
  uint32_t* crumbs = (uint32_t*)(ws + OFF_CRUMBS);
  uint32_t* hmap = (uint32_t*)(ws + OFF_HMAP);
  uint32_t* send = (uint32_t*)(ws + OFF_SEND);
  int* bestp = (int*)(ws + OFF_MISC);
  int* path = (int*)d_out;  // first SEQ_LEN elements: int32 best path
  float* score = (out_size > SEQ_LEN) ? ((float*)d_out + SEQ_LEN)
                                      : (float*)(ws + OFF_MISC + 16);

  k1_chunk_mats<<<NBLK, CPB, 0, stream>>>(em, tr, chunkM);
  k2_scan<<<1, K2_THREADS, 0, stream>>>(chunkM, alpha_in, bestp, score);
  k3_rescan<<<NBLK, CPB, 0, stream>>>(em, tr, alpha_in, crumbs, hmap);
  k4_backscan<<<1, K2_THREADS, 0, stream>>>(hmap, bestp, send);
  k5_writepath<<<NBLK, CPB, 0, stream>>>(crumbs, send, path);
}